// SlotGate_33019708571691
// MI455X (gfx1250) — compile-verified
//
#include <hip/hip_runtime.h>
#include <math.h>

typedef __attribute__((ext_vector_type(2))) float v2f;
typedef __attribute__((ext_vector_type(8))) float v8f;

#define D 256
#define NSLOT 4

// ---------------------------------------------------------------------------
// Phase 1: ragged embedding gather + sum.
// One block per batch row, thread d = embedding dim. Index addresses are
// wave-uniform (b, l uniform) -> scalar loads; row reads are fully coalesced.
// Writes t_q and t_sv = t_s + t_v to workspace.
// ---------------------------------------------------------------------------
__global__ __launch_bounds__(D) void gather_sum_kernel(
    const int* __restrict__ acts_q, const int* __restrict__ acts_s,
    const int* __restrict__ acts_v, const float* __restrict__ emb,
    float* __restrict__ t_q, float* __restrict__ t_sv, int L) {
  const int b = blockIdx.x;
  const int d = threadIdx.x;
  const int* iq = acts_q + (size_t)b * L;
  const int* is = acts_s + (size_t)b * L;
  const int* iv = acts_v + (size_t)b * L;
  float aq = 0.f, as = 0.f, av = 0.f;
  for (int l = 0; l < L; ++l) {
    aq += emb[(size_t)iq[l] * D + d];
    as += emb[(size_t)is[l] * D + d];
    av += emb[(size_t)iv[l] * D + d];
  }
  t_q[(size_t)b * D + d]  = aq;
  t_sv[(size_t)b * D + d] = as + av;
}

__device__ __forceinline__ float sigmoidf_(float x) {
  return 1.0f / (1.0f + __expf(-x));
}

// ---------------------------------------------------------------------------
// Phase 2 (fused): per-wave 16x16 tile of BOTH G = T @ W^T matmuls via
// V_WMMA_F32_16X16X4_F32, then the sigmoid-gate epilogue applied directly
// from the C-fragment registers (no g_q/g_sv round trip through memory).
//
// g[b][i] = sum_j w[i][j] * t[b][j]  ->  A[m][k]=t[b][j], B[k][n]=w[n][k].
//
// A 16x4 f32 layout: lane holds row M=lane&15; lanes 0-15 -> K={0,1},
// lanes 16-31 -> K={2,3}. B mirrored on columns. C/D 16x16 f32 layout:
// VGPR r: lanes 0-15 -> M=r, lanes 16-31 -> M=r+8; N=lane&15.
// ---------------------------------------------------------------------------
__global__ __launch_bounds__(32) void wmma_gate_kernel(
    const float* __restrict__ t_q, const float* __restrict__ t_sv,
    const float* __restrict__ w1, const float* __restrict__ w2,
    const int* __restrict__ slot_names, const float* __restrict__ emb,
    float* __restrict__ out) {
  const int lane  = threadIdx.x;          // 0..31 (wave32)
  const int mtile = blockIdx.x;           // BSZ/16 tiles
  const int ntile = blockIdx.y;           // D/16 tiles
  const int mn    = lane & 15;
  const int koff  = (lane >> 4) * 2;      // lanes 0-15: K+0/1, lanes 16-31: K+2/3

  const float* tq_row  = t_q  + (size_t)(mtile * 16 + mn) * D;
  const float* tsv_row = t_sv + (size_t)(mtile * 16 + mn) * D;
  const float* w1_row  = w1   + (size_t)(ntile * 16 + mn) * D;  // B[k][n]=w1[n][k]
  const float* w2_row  = w2   + (size_t)(ntile * 16 + mn) * D;

  v8f accq  = {};
  v8f accsv = {};
#pragma unroll 8
  for (int k = 0; k < D; k += 4) {
    v2f aq  = *(const v2f*)(tq_row  + k + koff);
    v2f asv = *(const v2f*)(tsv_row + k + koff);
    v2f bq  = *(const v2f*)(w1_row  + k + koff);
    v2f bsv = *(const v2f*)(w2_row  + k + koff);
    accq  = __builtin_amdgcn_wmma_f32_16x16x4_f32(
        /*neg_a=*/false, aq,  /*neg_b=*/false, bq,  (short)0, accq,  false, false);
    accsv = __builtin_amdgcn_wmma_f32_16x16x4_f32(
        /*neg_a=*/false, asv, /*neg_b=*/false, bsv, (short)0, accsv, false, false);
  }

  // Gate epilogue straight out of the C fragments.
  const int n     = ntile * 16 + mn;
  const int mbase = mtile * 16 + (lane >> 4) * 8;
#pragma unroll
  for (int vr = 0; vr < 8; ++vr) {
    const int m   = mbase + vr;
    const float gq  = accq[vr];
    const float gsv = accsv[vr];
    const int* sn = slot_names + (size_t)m * NSLOT;
    float* orow   = out + (size_t)m * NSLOT * D + n;
#pragma unroll
    for (int s = 0; s < NSLOT; ++s) {
      const float c = emb[(size_t)sn[s] * D + n];
      orow[(size_t)s * D] = c + sigmoidf_(c * gq) + sigmoidf_(c * gsv);
    }
  }
}

extern "C" void kernel_launch(void* const* d_in, const int* in_sizes, int n_in,
                              void* d_out, int out_size, void* d_ws, size_t ws_size,
                              hipStream_t stream) {
  const int*   acts_q     = (const int*)d_in[0];
  const int*   acts_s     = (const int*)d_in[1];
  const int*   acts_v     = (const int*)d_in[2];
  const int*   slot_names = (const int*)d_in[3];
  const float* emb        = (const float*)d_in[4];
  const float* w1         = (const float*)d_in[5];
  const float* w2         = (const float*)d_in[6];
  float*       out        = (float*)d_out;

  const int BSZ = in_sizes[3] / NSLOT;   // 4096
  const int L   = in_sizes[0] / BSZ;     // 50

  float* t_q  = (float*)d_ws;                       // BSZ*D f32
  float* t_sv = t_q + (size_t)BSZ * D;              // BSZ*D f32 (8 MB total)

  gather_sum_kernel<<<BSZ, D, 0, stream>>>(acts_q, acts_s, acts_v, emb,
                                           t_q, t_sv, L);

  dim3 grid(BSZ / 16, D / 16);
  wmma_gate_kernel<<<grid, 32, 0, stream>>>(t_q, t_sv, w1, w2,
                                            slot_names, emb, out);
}